// MDLSTM_42039139894314
// MI455X (gfx1250) — compile-verified
//
#include <hip/hip_runtime.h>
#include <hip/hip_bf16.h>
#include <math.h>

// MDLSTM wavefront implementation for gfx1250 (MI455X).
// D1=D2=64, B=32, I=64, O=128. 5 gate GEMMs fused per cell via
// V_WMMA_F32_16X16X4_F32 (exact fp32, matches reference math), wavefront
// recurrence over 127 anti-diagonals, one kernel launch per diagonal.
//
// d_ws usage: full-grid h buffer = 64*64*32*128 floats = 64 MB.

typedef float v2f __attribute__((ext_vector_type(2)));
typedef float v8f __attribute__((ext_vector_type(8)));

#define D1 64
#define D2 64
#define BSZ 32
#define ID 64
#define OD 128
#define NCOL 640            // 5 gates * 128 columns
#define CELL_ELEMS (BSZ * OD)   // 4096

__device__ __forceinline__ float sigf(float v) {
    return 1.0f / (1.0f + expf(-v));
}

__global__ void mdlstm_diag(const float* __restrict__ x,
                            const float* __restrict__ wf,  const float* __restrict__ uf,
                            const float* __restrict__ biasf,
                            const float* __restrict__ wi,  const float* __restrict__ ui,
                            const float* __restrict__ biasi,
                            const float* __restrict__ wo,  const float* __restrict__ uo,
                            const float* __restrict__ biaso,
                            const float* __restrict__ wc,  const float* __restrict__ uc,
                            const float* __restrict__ biasc,
                            float* __restrict__ s_out, float* __restrict__ hbuf,
                            int d, int lo) {
    extern __shared__ float Glds[];   // [32][640] gate pre-activations, 80 KB

    const int i = lo + (int)blockIdx.x;  // row in grid
    const int j = d - i;                 // col in grid

    const int tid  = threadIdx.x;
    const int lane = tid & 31;
    const int w    = tid >> 5;           // wave id 0..7
    const int lm   = lane & 15;          // M / N sub-index within tile
    const int klo  = 2 * (lane >> 4);    // K sub-offset for A/B operands
    const int rhi  = 8 * (lane >> 4);    // row offset for C/D halves

    const float* xc = x + (size_t)((i * D2 + j) * BSZ) * ID;  // this cell's (32,64) slab

    // ---------------- Phase 1: gates = bias + xj @ W via fp32 WMMA ----------------
    // 640 columns split over 8 waves -> 5 col-tiles of 16 per wave, 2 M-tiles (M=32).
    for (int t = 0; t < 5; ++t) {
        const int C  = (w * 5 + t) * 16;   // global column 0..624
        const int g  = C >> 7;             // gate index 0..4 (f0,f1,i,o,c)
        const int o0 = C & 127;
        const int col = o0 + lm;           // this lane's output column within gate

        const float* Wg;
        float bv;
        switch (g) {
            case 0:  Wg = wf;          bv = biasf[col];        break;
            case 1:  Wg = wf + ID*OD;  bv = biasf[OD + col];   break;
            case 2:  Wg = wi;          bv = biasi[col];        break;
            case 3:  Wg = wo;          bv = biaso[col];        break;
            default: Wg = wc;          bv = biasc[col];        break;
        }

        for (int m = 0; m < 2; ++m) {
            v8f acc = {bv, bv, bv, bv, bv, bv, bv, bv};
            const float* xrow = xc + (m * 16 + lm) * ID + klo;  // (M=m*16+lm, K=klo..)
#pragma unroll
            for (int k0 = 0; k0 < ID; k0 += 4) {
                // A: 16x4 f32 tile of x. lane holds (M=lm, K=k0+klo+r)
                v2f a = *(const v2f*)(xrow + k0);
                // B: 4x16 f32 tile of Wg. lane holds (K=k0+klo+r, N=col)
                const float* bp = Wg + (size_t)(k0 + klo) * OD + col;
                v2f b = {bp[0], bp[OD]};
                acc = __builtin_amdgcn_wmma_f32_16x16x4_f32(
                    false, a, false, b, (short)0, acc, false, false);
            }
            // C/D layout: vgpr r -> row (m*16 + r + rhi), col (C + lm)
#pragma unroll
            for (int r = 0; r < 8; ++r) {
                Glds[(m * 16 + r + rhi) * NCOL + C + lm] = acc[r];
            }
        }
    }

    __syncthreads();

    // ---------------- Phase 2: elementwise MDLSTM recurrence ----------------
    const size_t cell = (size_t)(i * D2 + j) * CELL_ELEMS;
    const bool has_up   = (i > 0);
    const bool has_left = (j > 0);

    for (int t = 0; t < 16; ++t) {
        const int e  = tid + 256 * t;    // 0..4095
        const int b  = e >> 7;           // batch 0..31
        const int o  = e & 127;          // output channel
        const int bo = b * OD + o;

        const float* gr = Glds + b * NCOL;
        const float gf0 = gr[o];
        const float gf1 = gr[OD + o];
        const float gi  = gr[2 * OD + o];
        const float go  = gr[3 * OD + o];
        const float gc  = gr[4 * OD + o];

        float s_up = 0.f, h_up = 0.f, s_lf = 0.f, h_lf = 0.f;
        if (has_up) {
            const size_t n = cell - (size_t)D2 * CELL_ELEMS + bo;
            s_up = s_out[n];
            h_up = hbuf[n];
        }
        if (has_left) {
            const size_t n = cell - CELL_ELEMS + bo;
            s_lf = s_out[n];
            h_lf = hbuf[n];
        }

        // uf: (2,2,O) -> uf[(l*2+k)*O + o]; k=0 up-neighbor, k=1 left-neighbor
        const float f0 = sigf(gf0 + h_up * uf[o]          + h_lf * uf[OD + o]);
        const float f1 = sigf(gf1 + h_up * uf[2 * OD + o] + h_lf * uf[3 * OD + o]);
        const float ig = sigf(gi  + h_up * ui[o] + h_lf * ui[OD + o]);
        const float og = sigf(go  + h_up * uo[o] + h_lf * uo[OD + o]);
        const float cg = sigf(gc  + h_up * uc[o] + h_lf * uc[OD + o]);

        const float s = ig * cg + f0 * s_up + f1 * s_lf;
        const float h = og * tanhf(s);

        s_out[cell + bo] = s;
        hbuf[cell + bo]  = h;
    }
}

extern "C" void kernel_launch(void* const* d_in, const int* in_sizes, int n_in,
                              void* d_out, int out_size, void* d_ws, size_t ws_size,
                              hipStream_t stream) {
    const float* x     = (const float*)d_in[0];
    const float* wf    = (const float*)d_in[1];
    const float* uf    = (const float*)d_in[2];
    const float* biasf = (const float*)d_in[3];
    const float* wi    = (const float*)d_in[4];
    const float* ui    = (const float*)d_in[5];
    const float* biasi = (const float*)d_in[6];
    const float* wo    = (const float*)d_in[7];
    const float* uo    = (const float*)d_in[8];
    const float* biaso = (const float*)d_in[9];
    const float* wc    = (const float*)d_in[10];
    const float* uc    = (const float*)d_in[11];
    const float* biasc = (const float*)d_in[12];

    float* s_out = (float*)d_out;
    float* hbuf  = (float*)d_ws;   // 64 MB full-grid h buffer

    const size_t lds_bytes = (size_t)BSZ * NCOL * sizeof(float);  // 80 KB

    // 127 wavefront steps; cells on a diagonal are independent blocks.
    for (int d = 0; d < D1 + D2 - 1; ++d) {
        const int lo = (d > D2 - 1) ? d - (D2 - 1) : 0;
        const int hi = (d < D1 - 1) ? d : D1 - 1;
        const int ncells = hi - lo + 1;
        mdlstm_diag<<<ncells, 256, lds_bytes, stream>>>(
            x, wf, uf, biasf, wi, ui, biasi, wo, uo, biaso, wc, uc, biasc,
            s_out, hbuf, d, lo);
    }
}